// IntersectionLoss_42949672960619
// MI455X (gfx1250) — compile-verified
//
#include <hip/hip_runtime.h>
#include <hip/hip_bf16.h>

typedef __attribute__((ext_vector_type(2))) float v2f;
typedef __attribute__((ext_vector_type(4))) float v4f;
typedef __attribute__((ext_vector_type(8))) float v8f;

#define K_RADIUS   1.0f
#define K_SIGMA    2.5f
#define K_EPSILON  1e-12f

// -log2(e) / sigma : exp(-x/sigma) == exp2(x * K_EXPSCALE)
#define K_EXPSCALE (-1.44269504088896340736f / K_SIGMA)
// ln(2): v_log_f32 returns log2(x); log(x) = log2(x) * LN2
#define K_LN2      0.69314718055994530942f

// ---------------------------------------------------------------------------
// Zero the per-batch accumulators (sum_d[N], sum_mask[N]) each launch.
// ---------------------------------------------------------------------------
__global__ void il_zero_kernel(float* __restrict__ ws, int count) {
    int i = blockIdx.x * blockDim.x + threadIdx.x;
    if (i < count) ws[i] = 0.0f;
}

// ---------------------------------------------------------------------------
// Prep: pack t1 points into the exact WMMA B-operand element order:
//   prep[i] = (-2x, -2y, -2z, ||p||^2)      (N*L1 points, 16B-aligned v4f)
// Lane lrow / half h of the consumer then needs elements {2h, 2h+1} of
// point j+lrow == one aligned b64 load, no selection, no divergence.
// ---------------------------------------------------------------------------
__global__ void il_prep_kernel(const float* __restrict__ t1,
                               float* __restrict__ prep, int total) {
    int i = blockIdx.x * blockDim.x + threadIdx.x;
    if (i >= total) return;
    const float x = t1[(size_t)i * 3 + 0];
    const float y = t1[(size_t)i * 3 + 1];
    const float z = t1[(size_t)i * 3 + 2];
    v4f o;
    o.x = -2.0f * x;
    o.y = -2.0f * y;
    o.z = -2.0f * z;
    o.w = fmaf(x, x, fmaf(y, y, z * z));
    *(v4f*)(prep + (size_t)i * 4) = o;   // 16B aligned
}

// ---------------------------------------------------------------------------
// Shared epilogue: 16-lane row reduction + log/clamp + masked atomics.
// ---------------------------------------------------------------------------
__device__ __forceinline__ void il_epilogue(
    float acc[8], const float* __restrict__ mask2, float* __restrict__ sums,
    int N, int L2, int n, int m0, int half, int lrow)
{
    // xor masks 1,2,4,8 never cross the half boundary on wave32.
#pragma unroll
    for (int r = 0; r < 8; ++r) {
        float s = acc[r];
        s += __shfl_xor(s, 1, 32);
        s += __shfl_xor(s, 2, 32);
        s += __shfl_xor(s, 4, 32);
        s += __shfl_xor(s, 8, 32);
        acc[r] = s;   // full sum for row (r + 8*half)
    }
    if (lrow == 0) {
        float sd = 0.0f, sm = 0.0f;
#pragma unroll
        for (int r = 0; r < 8; ++r) {
            const int row = m0 + r + 8 * half;
            // log(x) = v_log_f32(x) * ln2 ; acc >= eps > 0 so raw log is safe
            float dd = K_RADIUS +
                       (K_SIGMA * K_LN2) * __builtin_amdgcn_logf(acc[r] + K_EPSILON);
            dd = dd < 0.0f ? 0.0f : dd;
            const float m2 = mask2[(size_t)n * L2 + row];
            sd += dd * m2;
            sm += m2;
        }
        atomicAdd(&sums[n],     sd);
        atomicAdd(&sums[N + n], sm);
    }
}

// ---------------------------------------------------------------------------
// A-tile + C-base setup (shared by both main variants).
//   A (16x4, M=t2 rows) : (x2, y2, z2, 1)
//   C (16x16)           : ||t2_M||^2 broadcast along rows
// ---------------------------------------------------------------------------
__device__ __forceinline__ void il_setup_a(
    const float* __restrict__ t2, int n, int L2, int m0, int half, int lrow,
    v2f& a, v8f& cbase)
{
    const float* t2p = t2 + ((size_t)n * L2 + m0) * 3;
    const float x2 = t2p[lrow * 3 + 0];
    const float y2 = t2p[lrow * 3 + 1];
    const float z2 = t2p[lrow * 3 + 2];
    a.x = half ? z2   : x2;   // K = 2*half + 0
    a.y = half ? 1.0f : y2;   // K = 2*half + 1
    const float mynorm = fmaf(x2, x2, fmaf(y2, y2, z2 * z2));
#pragma unroll
    for (int r = 0; r < 8; ++r)
        cbase[r] = __shfl(mynorm, r + 8 * half, 32);
}

// ---------------------------------------------------------------------------
// Main kernel (prep path): one wave32 per 16-row t2 tile.
//   D[M][N] = ||t2_M - t1_N||^2 via V_WMMA_F32_16X16X4_F32,
//   B operand comes straight from the packed prep array (aligned b64).
// Inner loop per 256 pairs: 1 b64 + 1 b32 load, 1 WMMA, 4 pk_mul,
// 8 raw v_exp_f32 (TRANS floor), 4 pk_fma.
// ---------------------------------------------------------------------------
__global__ __launch_bounds__(256) void il_main_prep_kernel(
    const float* __restrict__ prep,   // (N, L1, 4) packed B elements
    const float* __restrict__ t2,     // (N, L2, 3)
    const float* __restrict__ mask1,  // (N, L1)
    const float* __restrict__ mask2,  // (N, L2)
    float* __restrict__ sums,
    int N, int L1, int L2)
{
    const int lane = threadIdx.x & 31;
    const int half = lane >> 4;
    const int lrow = lane & 15;

    const int w = blockIdx.x * (blockDim.x >> 5) + (threadIdx.x >> 5);
    const int tilesPerBatch = L2 >> 4;
    if (w >= N * tilesPerBatch) return;   // wave-uniform -> EXEC stays all-1s
    const int n  = w / tilesPerBatch;
    const int m0 = (w % tilesPerBatch) << 4;

    v2f a; v8f cbase;
    il_setup_a(t2, n, L2, m0, half, lrow, a, cbase);

    float acc[8];
#pragma unroll
    for (int r = 0; r < 8; ++r) acc[r] = 0.0f;

    // per-lane base: element {2*half} of point (j + lrow)
    const float* pb  = prep + ((size_t)n * L1 + lrow) * 4 + 2 * half;
    const float* m1b = mask1 + (size_t)n * L1 + lrow;

#pragma unroll 4
    for (int j = 0; j < L1; j += 16) {
        const v2f  b  = *(const v2f*)(pb + (size_t)j * 4);   // aligned b64
        const float m1 = m1b[j];

        v8f d = __builtin_amdgcn_wmma_f32_16x16x4_f32(
            false, a, false, b, (short)0, cbase, false, false);

#pragma unroll
        for (int r = 0; r < 8; ++r)
            acc[r] += __builtin_amdgcn_exp2f(d[r] * K_EXPSCALE) * m1;
    }

    il_epilogue(acc, mask2, sums, N, L2, n, m0, half, lrow);
}

// ---------------------------------------------------------------------------
// Fallback (no workspace for prep): build B in-loop, branchlessly
// (ternaries -> v_cndmask, no exec divergence).
// ---------------------------------------------------------------------------
__global__ __launch_bounds__(256) void il_main_fused_kernel(
    const float* __restrict__ t1,     // (N, L1, 3)
    const float* __restrict__ t2,     // (N, L2, 3)
    const float* __restrict__ mask1,  // (N, L1)
    const float* __restrict__ mask2,  // (N, L2)
    float* __restrict__ sums,
    int N, int L1, int L2)
{
    const int lane = threadIdx.x & 31;
    const int half = lane >> 4;
    const int lrow = lane & 15;

    const int w = blockIdx.x * (blockDim.x >> 5) + (threadIdx.x >> 5);
    const int tilesPerBatch = L2 >> 4;
    if (w >= N * tilesPerBatch) return;
    const int n  = w / tilesPerBatch;
    const int m0 = (w % tilesPerBatch) << 4;

    v2f a; v8f cbase;
    il_setup_a(t2, n, L2, m0, half, lrow, a, cbase);

    float acc[8];
#pragma unroll
    for (int r = 0; r < 8; ++r) acc[r] = 0.0f;

    const float* t1b = t1 + ((size_t)n * L1 + lrow) * 3;
    const float* m1b = mask1 + (size_t)n * L1 + lrow;

#pragma unroll 2
    for (int j = 0; j < L1; j += 16) {
        const float* p = t1b + (size_t)j * 3;
        const float x1 = p[0], y1 = p[1], z1 = p[2];
        const float n1 = fmaf(x1, x1, fmaf(y1, y1, z1 * z1));
        v2f b;
        b.x = -2.0f * (half ? z1 : x1);          // v_cndmask + mul
        b.y = half ? n1 : (-2.0f * y1);          // v_cndmask

        const float m1 = m1b[j];

        v8f d = __builtin_amdgcn_wmma_f32_16x16x4_f32(
            false, a, false, b, (short)0, cbase, false, false);

#pragma unroll
        for (int r = 0; r < 8; ++r)
            acc[r] += __builtin_amdgcn_exp2f(d[r] * K_EXPSCALE) * m1;
    }

    il_epilogue(acc, mask2, sums, N, L2, n, m0, half, lrow);
}

// ---------------------------------------------------------------------------
// Finalize: out[n] = sum_d[n] / sum_mask[n]
// ---------------------------------------------------------------------------
__global__ void il_finalize_kernel(const float* __restrict__ sums,
                                   float* __restrict__ out, int N) {
    int n = blockIdx.x * blockDim.x + threadIdx.x;
    if (n < N) out[n] = sums[n] / sums[N + n];
}

// ---------------------------------------------------------------------------
extern "C" void kernel_launch(void* const* d_in, const int* in_sizes, int n_in,
                              void* d_out, int out_size, void* d_ws, size_t ws_size,
                              hipStream_t stream) {
    const float* t1    = (const float*)d_in[0];  // (N, L1, 3)
    const float* t2    = (const float*)d_in[1];  // (N, L2, 3)
    const float* mask1 = (const float*)d_in[2];  // (N, L1)
    const float* mask2 = (const float*)d_in[3];  // (N, L2)
    float* out = (float*)d_out;                  // (N,)
    float* ws  = (float*)d_ws;

    const int N  = out_size;
    const int L1 = in_sizes[2] / N;
    const int L2 = in_sizes[3] / N;

    // ws layout: [sums: 2N floats][pad to 64][prep: N*L1*4 floats, 16B aligned]
    float* sums = ws;
    const size_t prepOff   = ((size_t)(2 * N) + 63) & ~(size_t)63;
    float* prep = ws + prepOff;
    const size_t needBytes = (prepOff + (size_t)N * L1 * 4) * sizeof(float);
    const bool usePrep = (ws_size >= needBytes);

    // 1) zero accumulators (ws is poisoned by the harness)
    il_zero_kernel<<<1, 64, 0, stream>>>(sums, 2 * N);

    // 2) main pass: one wave per 16-row t2 tile
    const int tilesPerBatch = L2 >> 4;
    const int totalWaves = N * tilesPerBatch;
    const int threads = 256;                        // 8 waves / block
    const int blocks  = (totalWaves * 32 + threads - 1) / threads;

    if (usePrep) {
        const int total = N * L1;
        il_prep_kernel<<<(total + 255) / 256, 256, 0, stream>>>(t1, prep, total);
        il_main_prep_kernel<<<blocks, threads, 0, stream>>>(
            prep, t2, mask1, mask2, sums, N, L1, L2);
    } else {
        il_main_fused_kernel<<<blocks, threads, 0, stream>>>(
            t1, t2, mask1, mask2, sums, N, L1, L2);
    }

    // 3) finalize masked mean
    il_finalize_kernel<<<1, 64, 0, stream>>>(sums, out, N);
}